// MyGCNNModel_65008624993069
// MI455X (gfx1250) — compile-verified
//
#include <hip/hip_runtime.h>
#include <hip/hip_bf16.h>

typedef float v2f __attribute__((ext_vector_type(2)));
typedef float v4f __attribute__((ext_vector_type(4)));
typedef float v8f __attribute__((ext_vector_type(8)));

#define BATCH 4
#define NN    8192
#define CH    16

// ---------------------------------------------------------------------------
// Projection: hp[i, o] = sum_k h[i, k] * W[o, k] + bias[o]
// One wave per 16-row tile; K=16 handled as four V_WMMA_F32_16X16X4_F32 steps.
// ---------------------------------------------------------------------------
__global__ __launch_bounds__(256) void proj_kernel(
    const float* __restrict__ h, const float* __restrict__ W,
    const float* __restrict__ bias, float* __restrict__ hp)
{
  const int lane = threadIdx.x & 31;
  const int wave = threadIdx.x >> 5;
  const int rc   = lane & 15;   // A-row / B-col / C-col
  const int hi   = lane >> 4;   // K half select
  const size_t i0 = ((size_t)blockIdx.x * 8 + wave) * 16;  // flat over B*N rows

  v8f acc = {0.f, 0.f, 0.f, 0.f, 0.f, 0.f, 0.f, 0.f};
#pragma unroll
  for (int k = 0; k < CH; k += 4) {
    const int kk = k + hi * 2;
    v2f a = *(const v2f*)(h + (i0 + rc) * CH + kk);
    v2f bf;
    bf.x = W[rc * CH + kk];
    bf.y = W[rc * CH + kk + 1];
    acc = __builtin_amdgcn_wmma_f32_16x16x4_f32(
        false, a, false, bf, (short)0, acc, false, false);
  }

  const float bv = bias[rc];
#pragma unroll
  for (int e = 0; e < 8; ++e) {
    const int m = e + hi * 8;                 // C layout: vgpr e -> row e (+8 hi half)
    hp[(i0 + m) * CH + rc] = acc[e] + bv;
  }
}

// ---------------------------------------------------------------------------
// Aggregation: hout[b,i,:] = relu( (adj[b,i,:] @ hp[b,:,:]) / deg[b,i] )
// One workgroup per (batch, 16-row tile); 8 waves split K=8192.
// K-permuted dual-WMMA scheme: one global_load_b128 of adj per lane
// (adj[row, kb+hi*4 .. +3]) feeds TWO v_wmma_f32_16x16x4_f32 ops:
//   WMMA-1 slots {0,1}(hi=0) / {2,3}(hi=1) -> K = {kb,kb+1,kb+4,kb+5}
//   WMMA-2 -> K = {kb+2,kb+3,kb+6,kb+7}
// (valid because the K-sum is commutative; B rows fetched to match).
// Layer 0 fuses the degree row-sum from A-fragments already in registers.
// ---------------------------------------------------------------------------
__global__ __launch_bounds__(256) void agg_kernel(
    const float* __restrict__ adj, const float* __restrict__ hp,
    float* __restrict__ hout, float* __restrict__ deg, int layer0)
{
  __shared__ __align__(32) float cred[8 * 256];  // 8 partial 16x16 C tiles
  __shared__ float degl[16];

  const int tid  = threadIdx.x;
  const int lane = tid & 31;
  const int wave = tid >> 5;
  const int rc   = lane & 15;
  const int hi   = lane >> 4;
  const int b    = blockIdx.y;
  const int i0   = blockIdx.x * 16;

  const float* arow = adj + ((size_t)b * NN + i0 + rc) * NN;  // this lane's adj row
  const float* hb   = hp + (size_t)b * NN * CH;

  v8f  acc  = {0.f, 0.f, 0.f, 0.f, 0.f, 0.f, 0.f, 0.f};
  float dsum = 0.f;
  const int k0 = wave * (NN / 8);

  for (int k = k0; k < k0 + NN / 8; k += 32) {
    __builtin_prefetch(arow + k + 1024, 0, 0);   // speculative stream prefetch 4KB ahead
#pragma unroll
    for (int u = 0; u < 32; u += 8) {
      const int kb = k + u;
      const int kk = kb + hi * 4;
      v4f a = *(const v4f*)(arow + kk);          // adj[i0+rc, kk .. kk+3]
      v2f a0 = {a.x, a.y};
      v2f a1 = {a.z, a.w};
      v2f b0, b1;
      b0.x = hb[(size_t)(kk    ) * CH + rc];     // hp[kk,   rc]
      b0.y = hb[(size_t)(kk + 1) * CH + rc];
      b1.x = hb[(size_t)(kk + 2) * CH + rc];
      b1.y = hb[(size_t)(kk + 3) * CH + rc];
      acc = __builtin_amdgcn_wmma_f32_16x16x4_f32(
          false, a0, false, b0, (short)0, acc, false, false);
      acc = __builtin_amdgcn_wmma_f32_16x16x4_f32(
          false, a1, false, b1, (short)0, acc, false, false);
      dsum += a.x + a.y + a.z + a.w;             // fused degree partial (layer 0)
    }
  }

  if (tid < 16) degl[tid] = 0.f;
  *(v8f*)&cred[wave * 256 + lane * 8] = acc;     // 32B-aligned ds_store
  __syncthreads();
  if (layer0) atomicAdd(&degl[rc], dsum);        // ds_add_f32; lanes {r,r+16} x 8 waves
  __syncthreads();

  // Elementwise cross-wave reduction: thread t owns flat C element t.
  const int l = tid >> 3;
  const int e = tid & 7;
  const int m = e + ((l >> 4) << 3);
  const int n = l & 15;
  float s = 0.f;
#pragma unroll
  for (int w = 0; w < 8; ++w) s += cred[w * 256 + tid];

  float d;
  if (layer0) {
    d = degl[m];
    if (n == 0) deg[(size_t)b * NN + i0 + m] = d;
  } else {
    d = deg[(size_t)b * NN + i0 + m];
  }
  float v = s / d;
  v = v > 0.f ? v : 0.f;
  hout[((size_t)b * NN + i0 + m) * CH + n] = v;
}

// ---------------------------------------------------------------------------
// Head: out[b,o] = sum_i scrub(h[b,i]) * W_fl[o,i] + b_fl[o]
// ---------------------------------------------------------------------------
__global__ __launch_bounds__(256) void head_kernel(
    const float* __restrict__ h, const float* __restrict__ Wfl,
    const float* __restrict__ bfl, float* __restrict__ out)
{
  __shared__ float red[256];
  const int b = blockIdx.x >> 1;
  const int o = blockIdx.x & 1;
  const float* hb = h + (size_t)b * NN * CH;
  const float* wr = Wfl + (size_t)o * NN * CH;

  float p = 0.f;
  for (int i = threadIdx.x; i < NN * CH; i += 256) {
    float v = hb[i];
    v = (v != v) ? 0.f : v;                    // NaN scrub
    p = fmaf(v, wr[i], p);
  }
  red[threadIdx.x] = p;
  __syncthreads();
#pragma unroll
  for (int s = 128; s > 0; s >>= 1) {
    if (threadIdx.x < s) red[threadIdx.x] += red[threadIdx.x + s];
    __syncthreads();
  }
  if (threadIdx.x == 0) out[blockIdx.x] = red[0] + bfl[o];
}

// ---------------------------------------------------------------------------
extern "C" void kernel_launch(void* const* d_in, const int* in_sizes, int n_in,
                              void* d_out, int out_size, void* d_ws, size_t ws_size,
                              hipStream_t stream) {
  const float* node = (const float*)d_in[0];   // [B,N,16]
  const float* adj  = (const float*)d_in[1];   // [B,N,N]
  const float* Wp   = (const float*)d_in[2];   // [16,16]
  const float* bp   = (const float*)d_in[3];   // [16]
  const float* Wfl  = (const float*)d_in[4];   // [2, N*16]
  const float* bfl  = (const float*)d_in[5];   // [2]
  float* out = (float*)d_out;

  float* ws  = (float*)d_ws;
  float* hp  = ws;                              // [B,N,16]
  float* hb  = ws + (size_t)BATCH * NN * CH;    // [B,N,16]
  float* deg = ws + 2 * (size_t)BATCH * NN * CH;// [B,N]

  const dim3 gProj(BATCH * NN / (16 * 8));      // 2048 tiles, 8 per block
  const dim3 gAgg(NN / 16, BATCH);              // 512 x 4 workgroups

  const float* cur = node;
  for (int layer = 0; layer < 3; ++layer) {
    proj_kernel<<<gProj, 256, 0, stream>>>(cur, Wp, bp, hp);
    agg_kernel<<<gAgg, 256, 0, stream>>>(adj, hp, hb, deg, layer == 0 ? 1 : 0);
    cur = hb;
  }
  head_kernel<<<8, 256, 0, stream>>>(hb, Wfl, bfl, out);
}